// HyenaOperator_18854906429533
// MI455X (gfx1250) — compile-verified
//
#include <hip/hip_runtime.h>
#include <math.h>
#include <stdint.h>

typedef __attribute__((ext_vector_type(2))) float v2f;
typedef __attribute__((ext_vector_type(8))) float v8f;
typedef __attribute__((ext_vector_type(4))) unsigned int u32x4;
typedef __attribute__((ext_vector_type(4))) int i32x4;
typedef __attribute__((ext_vector_type(8))) int i32x8;

#define D_MODEL     1024
#define L_MAX       2048
#define BATCH       4
#define TOTAL_WIDTH 3072

// ---- Tensor Data Mover availability (device pass only; host pass parses
//      the fallback path). Arity differs: clang-22 (ROCm 7.2) = 5 args,
//      clang-23 (amdgpu-toolchain) = 6 args.
#if defined(__has_builtin)
#  if __has_builtin(__builtin_amdgcn_tensor_load_to_lds)
#    define HYENA_HAVE_TDM 1
#  endif
#endif
#ifndef HYENA_HAVE_TDM
#  define HYENA_HAVE_TDM 0
#endif

#if HYENA_HAVE_TDM
// 2-D tile load Global->LDS via TDM descriptor (D#), data_size = 4 bytes.
// tile_x/tile_y = tile dims (elements), stride_elems = row stride of tensor,
// pad_iv/pad_amt = encoded LDS padding (interval 2^(iv+1) dwords, amount amt+1).
__device__ __forceinline__ void tdm_load_2d(
    unsigned lds_addr, const void* gptr,
    unsigned tile_x, unsigned tile_y,
    unsigned tensor_x, unsigned tensor_y,
    unsigned long long stride_elems,
    unsigned pad_iv, unsigned pad_amt, bool pad_en)
{
    const unsigned long long ga = (unsigned long long)(uintptr_t)gptr;
    u32x4 g0;
    g0.x = 1u;                                        // count=1 (valid user D#)
    g0.y = lds_addr;                                  // lds_addr [63:32]
    g0.z = (unsigned)(ga & 0xFFFFFFFFu);              // global_addr [95:64]
    g0.w = (unsigned)((ga >> 32) & 0x01FFFFFFu)       // global_addr [120:96]
         | (2u << 30);                                // type=2 ("image")
    unsigned w0 = (2u << 16);                         // data_size=2 -> 4 bytes
    if (pad_en) w0 |= (1u << 20) | (pad_iv << 22) | (pad_amt << 25);
    i32x8 g1;
    g1[0] = (int)w0;                                  // mask=0, flags, padding
    g1[1] = (int)((tensor_x & 0xFFFFu) << 16);        // tensor_dim0[15:0]
    g1[2] = (int)((tensor_x >> 16) | ((tensor_y & 0xFFFFu) << 16));
    g1[3] = (int)((tensor_y >> 16) | (tile_x << 16)); // tile_dim0
    g1[4] = (int)(tile_y & 0xFFFFu);                  // tile_dim1, tile_dim2=0
    g1[5] = (int)(unsigned)(stride_elems & 0xFFFFFFFFull);     // dim0_stride lo
    g1[6] = (int)(unsigned)((stride_elems >> 32) & 0xFFFFull); // dim0_stride hi
    g1[7] = 0;
    const i32x4 gz = {0, 0, 0, 0};
#if __clang_major__ >= 23
    const i32x8 gz8 = {0, 0, 0, 0, 0, 0, 0, 0};
    __builtin_amdgcn_tensor_load_to_lds(g0, g1, gz, gz, gz8, 0);
#else
    __builtin_amdgcn_tensor_load_to_lds(g0, g1, gz, gz, 0);
#endif
}
__device__ __forceinline__ unsigned lds_off(const void* p) {
    return (unsigned)(unsigned long long)(uintptr_t)p;   // AS(3) offset bits
}
#endif

// ---------------------------------------------------------------------------
// GEMM: C = A(MxK) @ W(KxN) + bias, via V_WMMA_F32_16X16X4_F32.
// Block = 256 threads (8 waves), tile 64(M) x 64(N), K in 16-chunks,
// double-buffered LDS staged by the Tensor Data Mover (TENSORcnt-tracked).
// A tile lands at row stride 18 dwords, W tile at 68 (TDM pad => conflict-free).
// ---------------------------------------------------------------------------
template <bool TRANSPOSE_OUT>
__global__ __launch_bounds__(256) void gemm_wmma_f32(
    const float* __restrict__ A, const float* __restrict__ W,
    const float* __restrict__ bias, float* __restrict__ C,
    int M, int N, int K)
{
    __shared__ float As[2][64 * 18];   // As[m][k], stride 18
    __shared__ float Ws[2][16 * 68];   // Ws[k][n], stride 68

    const int tid  = threadIdx.x;
    const int wid  = tid >> 5;
    const int lane = tid & 31;
    const int h    = lane >> 4;
    const int r    = lane & 15;

    const int m0 = blockIdx.y * 64;
    const int n0 = blockIdx.x * 64;
    const int mr = (wid >> 2) * 32;
    const int nc = (wid & 3) * 16;

    v8f acc0 = {};
    v8f acc1 = {};

    auto stage = [&](int k0, int bsel) {
#if HYENA_HAVE_TDM
        if (wid == 0) {
            // A tile: 64 rows x 16 k; pad 2 dwords every 16 -> LDS stride 18.
            tdm_load_2d(lds_off(&As[bsel][0]), &A[(size_t)m0 * K + k0],
                        /*tile_x=*/16, /*tile_y=*/64,
                        /*tensor_x=*/(unsigned)(K - k0), /*tensor_y=*/(unsigned)(M - m0),
                        (unsigned long long)K, /*iv:16dw*/3, /*amt:2dw*/1, true);
            // W tile: 16 rows x 64 n; pad 4 dwords every 64 -> LDS stride 68.
            tdm_load_2d(lds_off(&Ws[bsel][0]), &W[(size_t)k0 * N + n0],
                        /*tile_x=*/64, /*tile_y=*/16,
                        /*tensor_x=*/(unsigned)(N - n0), /*tensor_y=*/(unsigned)(K - k0),
                        (unsigned long long)N, /*iv:64dw*/5, /*amt:4dw*/3, true);
        }
#else
        for (int e = tid; e < 64 * 16; e += 256) {
            int m = e >> 4, k = e & 15;
            As[bsel][m * 18 + k] = A[(size_t)(m0 + m) * K + (k0 + k)];
        }
        for (int e = tid; e < 16 * 64; e += 256) {
            int k = e >> 6, n = e & 63;
            Ws[bsel][k * 68 + n] = W[(size_t)(k0 + k) * N + (n0 + n)];
        }
        if (k0 + 16 < K)
            __builtin_prefetch(&A[(size_t)(m0 + (tid >> 2)) * K + k0 + 16], 0, 1);
#endif
    };

    stage(0, 0);
    int buf = 0;
    for (int k0 = 0; k0 < K; k0 += 16) {
        const bool has_next = (k0 + 16 < K);
        if (has_next) stage(k0 + 16, buf ^ 1);
#if HYENA_HAVE_TDM
        if (has_next) __builtin_amdgcn_s_wait_tensorcnt(2);  // current chunk done
        else          __builtin_amdgcn_s_wait_tensorcnt(0);
#endif
        __syncthreads();
        const float* as = As[buf];
        const float* ws = Ws[buf];
#pragma unroll
        for (int s = 0; s < 4; ++s) {
            const int kk = 4 * s + 2 * h;
            v2f bfr;
            bfr.x = ws[kk * 68 + nc + r];
            bfr.y = ws[(kk + 1) * 68 + nc + r];
            v2f a0, a1;
            a0.x = as[(mr + r) * 18 + kk];
            a0.y = as[(mr + r) * 18 + kk + 1];
            a1.x = as[(mr + 16 + r) * 18 + kk];
            a1.y = as[(mr + 16 + r) * 18 + kk + 1];
            acc0 = __builtin_amdgcn_wmma_f32_16x16x4_f32(false, a0, false, bfr,
                                                         (short)0, acc0, false, false);
            acc1 = __builtin_amdgcn_wmma_f32_16x16x4_f32(false, a1, false, bfr,
                                                         (short)0, acc1, false, false);
        }
        __syncthreads();
        buf ^= 1;
    }

    const int   cn = n0 + nc + r;
    const float bv = bias[cn];
#pragma unroll
    for (int j = 0; j < 8; ++j) {
        const int row = j + 8 * h;
        const int ma  = m0 + mr + row;
        const int mb  = m0 + mr + 16 + row;
        const float va = acc0[j] + bv;
        const float vb = acc1[j] + bv;
        if (TRANSPOSE_OUT) {
            const int ba = ma >> 11, la = ma & (L_MAX - 1);
            const int bb = mb >> 11, lb = mb & (L_MAX - 1);
            C[((size_t)ba * N + cn) * L_MAX + la] = va;
            C[((size_t)bb * N + cn) * L_MAX + lb] = vb;
        } else {
            C[(size_t)ma * N + cn] = va;
            C[(size_t)mb * N + cn] = vb;
        }
    }
}

// ---------------------------------------------------------------------------
// Depthwise causal short conv (k=3) + bias on all 3 branches, then gate.
// ---------------------------------------------------------------------------
__global__ __launch_bounds__(256) void shortconv_gate(
    const float* __restrict__ up, const float* __restrict__ sw,
    const float* __restrict__ sb, float* __restrict__ x0,
    float* __restrict__ vg)
{
    const size_t idx = (size_t)blockIdx.x * blockDim.x + threadIdx.x;
    const size_t total = (size_t)BATCH * D_MODEL * L_MAX;
    if (idx >= total) return;
    const int l = (int)(idx & (L_MAX - 1));
    const size_t bc = idx >> 11;
    const int c = (int)(bc & (D_MODEL - 1));
    const int b = (int)(bc >> 10);

    float out[3];
#pragma unroll
    for (int g = 0; g < 3; ++g) {
        const int ch = c + g * D_MODEL;
        const float* row = up + ((size_t)b * TOTAL_WIDTH + ch) * L_MAX;
        float acc = sb[ch];
#pragma unroll
        for (int j = 0; j < 3; ++j) {
            const int ll = l - 2 + j;
            const float x = (ll >= 0) ? row[ll] : 0.0f;
            acc += sw[ch * 3 + j] * x;
        }
        out[g] = acc;
    }
    x0[idx] = out[0];
    vg[idx] = out[2] * out[1];
}

// ---------------------------------------------------------------------------
// Implicit filter MLP + exponential modulation. One block per position l.
// ---------------------------------------------------------------------------
__global__ __launch_bounds__(256) void hyena_filter(
    const float* __restrict__ w0, const float* __restrict__ b0, const float* __restrict__ f0,
    const float* __restrict__ w1, const float* __restrict__ b1, const float* __restrict__ f1,
    const float* __restrict__ w2, const float* __restrict__ b2, const float* __restrict__ f2,
    const float* __restrict__ w3, float* __restrict__ kf)
{
    __shared__ float ha[64];
    __shared__ float hb[64];
    const int l   = blockIdx.x;
    const int tid = threadIdx.x;

    const float t     = (float)l / (float)(L_MAX - 1);
    const float wang  = 2.0f * 3.14159265358979323846f * (float)l / (float)L_MAX;
    const float phase = 1e-4f * wang;
    const float z0 = t, z1 = cosf(phase), z2 = -sinf(phase);

    if (tid < 64) {
        float s = z0 * w0[tid] + z1 * w0[64 + tid] + z2 * w0[128 + tid] + b0[tid];
        ha[tid] = sinf(f0[tid] * s);
    }
    __syncthreads();
    if (tid < 64) {
        float s = b1[tid];
        for (int i = 0; i < 64; ++i) s += ha[i] * w1[i * 64 + tid];
        hb[tid] = sinf(f1[tid] * s);
    }
    __syncthreads();
    if (tid < 64) {
        float s = b2[tid];
        for (int i = 0; i < 64; ++i) s += hb[i] * w2[i * 64 + tid];
        ha[tid] = sinf(f2[tid] * s);
    }
    __syncthreads();

    const float MIN_DECAY = -3.0701134573253940f;   // ln(1e-2)/1.5
    const float MAX_DECAY = -15.3505672866269710f;  // ln(1e-2)/0.3
    for (int c = tid; c < D_MODEL; c += 256) {
        float s = 0.0f;
        for (int i = 0; i < 64; ++i) s += ha[i] * w3[i * D_MODEL + c];
        const float delta = MIN_DECAY + (MAX_DECAY - MIN_DECAY) * (float)c / (float)(D_MODEL - 1);
        kf[(size_t)c * L_MAX + l] = s * expf(-t * fabsf(delta));
    }
}

// ---------------------------------------------------------------------------
// Long causal conv via blocked-Toeplitz WMMA; k/v rows staged by TDM.
// Epilogue fuses + v*filter_bias, * x0 gate, (b,c,l)->(b,l,c) transpose.
// ---------------------------------------------------------------------------
__global__ __launch_bounds__(256) void long_conv_wmma(
    const float* __restrict__ kf, const float* __restrict__ vg,
    const float* __restrict__ x0, const float* __restrict__ fbias,
    float* __restrict__ yin)
{
    __shared__ float kl[L_MAX];
    __shared__ float vl[L_MAX];
    const int tid = threadIdx.x;
    const int bc  = blockIdx.x;
    const int c   = bc & (D_MODEL - 1);
    const int b   = bc >> 10;

    const float* krow = kf + (size_t)c * L_MAX;
    const float* vrow = vg + (size_t)bc * L_MAX;
#if HYENA_HAVE_TDM
    if ((tid >> 5) == 0) {
        tdm_load_2d(lds_off(kl), krow, L_MAX, 1, L_MAX, 1,
                    (unsigned long long)L_MAX, 0, 0, false);
        tdm_load_2d(lds_off(vl), vrow, L_MAX, 1, L_MAX, 1,
                    (unsigned long long)L_MAX, 0, 0, false);
    }
    __builtin_amdgcn_s_wait_tensorcnt(0);
    __syncthreads();
#else
    for (int e = tid; e < L_MAX; e += 256) { kl[e] = krow[e]; vl[e] = vrow[e]; }
    __syncthreads();
#endif

    const int w    = tid >> 5;
    const int lane = tid & 31;
    const int h    = lane >> 4;
    const int r    = lane & 15;
    const int i_n  = 8 * r + w;        // interleaved output tile per column
    const int dmax = 8 * 15 + w;

    v8f acc = {};
    for (int d = 0; d <= dmax; ++d) {
        const bool colok = (i_n >= d);
        const int  vbase = 16 * (i_n - d);
        const int  abase = 16 * d + r;
#pragma unroll
        for (int s = 0; s < 4; ++s) {
            const int q = 4 * s + 2 * h;
            v2f a, bfr;
            const int ax = abase - q;
            a.x = (ax >= 0)     ? kl[ax]     : 0.0f;
            a.y = (ax - 1 >= 0) ? kl[ax - 1] : 0.0f;
            bfr.x = colok ? vl[vbase + q]     : 0.0f;
            bfr.y = colok ? vl[vbase + q + 1] : 0.0f;
            acc = __builtin_amdgcn_wmma_f32_16x16x4_f32(false, a, false, bfr,
                                                        (short)0, acc, false, false);
        }
    }

    const float fb = fbias[c];
    const float* x0row = x0 + (size_t)bc * L_MAX;
#pragma unroll
    for (int j = 0; j < 8; ++j) {
        const int t = 16 * i_n + j + 8 * h;
        float y = acc[j] + fb * vl[t];
        y *= x0row[t];
        yin[((size_t)b * L_MAX + t) * D_MODEL + c] = y;
    }
}

// ---------------------------------------------------------------------------
extern "C" void kernel_launch(void* const* d_in, const int* in_sizes, int n_in,
                              void* d_out, int out_size, void* d_ws, size_t ws_size,
                              hipStream_t stream)
{
    (void)in_sizes; (void)n_in; (void)out_size; (void)ws_size;
    const float* u           = (const float*)d_in[0];
    const float* in_w        = (const float*)d_in[1];
    const float* in_b        = (const float*)d_in[2];
    const float* short_w     = (const float*)d_in[3];
    const float* short_b     = (const float*)d_in[4];
    const float* mlp_w0      = (const float*)d_in[5];
    const float* mlp_b0      = (const float*)d_in[6];
    const float* freq0       = (const float*)d_in[7];
    const float* mlp_w1      = (const float*)d_in[8];
    const float* mlp_b1      = (const float*)d_in[9];
    const float* freq1       = (const float*)d_in[10];
    const float* mlp_w2      = (const float*)d_in[11];
    const float* mlp_b2      = (const float*)d_in[12];
    const float* freq2       = (const float*)d_in[13];
    const float* mlp_w3      = (const float*)d_in[14];
    const float* filter_bias = (const float*)d_in[15];
    const float* out_w       = (const float*)d_in[16];
    const float* out_b       = (const float*)d_in[17];
    float* out = (float*)d_out;

    // Workspace: up (100663296 B, reused as yin) | x0 | vg | kf
    char*  ws  = (char*)d_ws;
    float* up  = (float*)(ws);
    float* x0  = (float*)(ws + 100663296ull);
    float* vg  = (float*)(ws + 100663296ull + 33554432ull);
    float* kf  = (float*)(ws + 100663296ull + 2ull * 33554432ull);
    float* yin = up;

    const int M = BATCH * L_MAX;   // 8192

    {   // 1) in_proj -> channel-major up[b][c][l]
        dim3 grid(TOTAL_WIDTH / 64, M / 64);
        gemm_wmma_f32<true><<<grid, dim3(256), 0, stream>>>(
            u, in_w, in_b, up, M, TOTAL_WIDTH, D_MODEL);
    }
    {   // 2) short conv + gate
        const size_t total = (size_t)BATCH * D_MODEL * L_MAX;
        shortconv_gate<<<dim3((unsigned)((total + 255) / 256)), dim3(256), 0, stream>>>(
            up, short_w, short_b, x0, vg);
    }
    // 3) implicit filter -> kf[c][l]
    hyena_filter<<<dim3(L_MAX), dim3(256), 0, stream>>>(
        mlp_w0, mlp_b0, freq0, mlp_w1, mlp_b1, freq1,
        mlp_w2, mlp_b2, freq2, mlp_w3, kf);
    // 4) long conv (WMMA Toeplitz) -> yin[b][l][c]
    long_conv_wmma<<<dim3(BATCH * D_MODEL), dim3(256), 0, stream>>>(
        kf, vg, x0, filter_bias, yin);
    {   // 5) out_proj -> d_out
        dim3 grid(D_MODEL / 64, M / 64);
        gemm_wmma_f32<false><<<grid, dim3(256), 0, stream>>>(
            yin, out_w, out_b, out, M, D_MODEL, D_MODEL);
    }
}